// RepBEV_vit_60387240182396
// MI455X (gfx1250) — compile-verified
//
#include <hip/hip_runtime.h>

typedef float v2f __attribute__((ext_vector_type(2)));
typedef float v8f __attribute__((ext_vector_type(8)));

#define BB   4
#define NN   6
#define DD   41
#define HH   16
#define WW   44
#define CC   64
#define HW   704        // HH*WW
#define DHW  28864      // DD*HH*WW
#define NPTS 692736     // BB*NN*DHW
#define NXY  200
#define CELLS 40000     // 200*200
#define OUTF 10240000   // BB*CC*CELLS

__device__ __forceinline__ void inv3x3(const float m[9], float inv[9]) {
  float c00 =  m[4]*m[8] - m[5]*m[7];
  float c01 = -(m[3]*m[8] - m[5]*m[6]);
  float c02 =  m[3]*m[7] - m[4]*m[6];
  float id  = 1.0f / (m[0]*c00 + m[1]*c01 + m[2]*c02);
  inv[0] = c00*id;
  inv[1] = (m[2]*m[7] - m[1]*m[8])*id;
  inv[2] = (m[1]*m[5] - m[2]*m[4])*id;
  inv[3] = c01*id;
  inv[4] = (m[0]*m[8] - m[2]*m[6])*id;
  inv[5] = (m[2]*m[3] - m[0]*m[5])*id;
  inv[6] = c02*id;
  inv[7] = (m[1]*m[6] - m[0]*m[7])*id;
  inv[8] = (m[0]*m[4] - m[1]*m[3])*id;
}

// ---- kernel 0: zero the BEV grid (harness poisons d_out; we must re-zero per call)
__global__ void lss_zero(float4* __restrict__ out4) {
  int i = blockIdx.x * blockDim.x + threadIdx.x;   // grid sized exactly: OUTF/4 stores
  out4[i] = make_float4(0.f, 0.f, 0.f, 0.f);
}

// ---- kernel 1: per-(b,n) matrix setup. mats[t*32 + 0..15] = inv(ida), +16..31 = combine
__global__ void lss_setup(const float* __restrict__ rots, const float* __restrict__ trans,
                          const float* __restrict__ intrins, const float* __restrict__ post_rots,
                          const float* __restrict__ post_trans, float* __restrict__ mats) {
  int t = threadIdx.x;
  if (t >= BB * NN) return;
  float pr[9], pri[9], K[9], Ki[9], R[9];
#pragma unroll
  for (int i = 0; i < 9; ++i) { pr[i] = post_rots[t*9+i]; K[i] = intrins[t*9+i]; R[i] = rots[t*9+i]; }
  inv3x3(pr, pri);
  inv3x3(K, Ki);
  float p0 = post_trans[t*3+0], p1 = post_trans[t*3+1], p2 = post_trans[t*3+2];
  float* M = mats + t * 32;
#pragma unroll
  for (int r = 0; r < 3; ++r) {
    M[r*4+0] = pri[r*3+0]; M[r*4+1] = pri[r*3+1]; M[r*4+2] = pri[r*3+2];
    M[r*4+3] = -(pri[r*3+0]*p0 + pri[r*3+1]*p1 + pri[r*3+2]*p2);
  }
  M[12] = 0.f; M[13] = 0.f; M[14] = 0.f; M[15] = 1.f;
  float* Cm = M + 16;                                  // combine = [R*K^-1 | trans; 0 0 0 1]
#pragma unroll
  for (int r = 0; r < 3; ++r) {
#pragma unroll
    for (int c = 0; c < 3; ++c)
      Cm[r*4+c] = R[r*3+0]*Ki[0*3+c] + R[r*3+1]*Ki[1*3+c] + R[r*3+2]*Ki[2*3+c];
    Cm[r*4+3] = trans[t*3+r];
  }
  Cm[12] = 0.f; Cm[13] = 0.f; Cm[14] = 0.f; Cm[15] = 1.f;
}

// ---- kernel 2: geometry via V_WMMA_F32_16X16X4_F32. One wave = 16 points of one (b,n).
// D(16x16) = A(16x4: combine rows, padded) x B(4x16: homogeneous point vectors).
// 32-bit A 16x4 layout: VGPR0 = K0 (lanes 0-15) / K2 (lanes 16-31); VGPR1 = K1 / K3.
// D layout: lane n<16 holds column N=n; VGPR r holds row M=r -> lane n gets point n's x,y,z in D[0..2].
__global__ void lss_geom(const float* __restrict__ mats, int* __restrict__ cellOff) {
  int gt     = blockIdx.x * blockDim.x + threadIdx.x;
  int waveId = gt >> 5;
  int lane   = threadIdx.x & 31;
  int bn     = waveId / (DHW / 16);                    // 1804 waves per (b,n)
  int chunk  = waveId - bn * (DHW / 16);
  int pl     = lane & 15;
  int pt     = chunk * 16 + pl;
  int di     = pt / HW;
  int r      = pt - di * HW;
  int h      = r / WW;
  int w      = r - h * WW;
  // frustum: xs in [0,703] over 44, ys in [0,255] over 16, depth bins 4..44
  float xs  = (float)w * (703.0f / 43.0f);
  float ys  = (float)h * 17.0f;
  float dep = 4.0f + (float)di;
  const float* Mi = mats + bn * 32;                    // inv(ida)
  float u = Mi[0]*xs + Mi[1]*ys + Mi[2]*dep  + Mi[3];
  float v = Mi[4]*xs + Mi[5]*ys + Mi[6]*dep  + Mi[7];
  float z = Mi[8]*xs + Mi[9]*ys + Mi[10]*dep + Mi[11];
  const float* Cm = Mi + 16;                           // combine
  bool hi = lane >= 16;
  // A operand: rows 0..3 = combine, rows 4..15 = 0; low lanes carry K0/K1, high lanes K2/K3
  float a0 = (pl < 4) ? Cm[pl*4 + (hi ? 2 : 0)] : 0.0f;
  float a1 = (pl < 4) ? Cm[pl*4 + (hi ? 3 : 1)] : 0.0f;
  // B operand: column pl = [u*z, v*z, z, 1]; low lanes carry K0/K1 rows, high lanes K2/K3 rows
  float b0 = hi ? z    : u * z;
  float b1 = hi ? 1.0f : v * z;
  v2f A;  A[0] = a0;  A[1] = a1;
  v2f Bv; Bv[0] = b0; Bv[1] = b1;
  v8f Cz = {};
  v8f Dm = __builtin_amdgcn_wmma_f32_16x16x4_f32(false, A, false, Bv, (short)0, Cz, false, false);
  if (lane < 16) {
    float gx = Dm[0], gy = Dm[1], gz = Dm[2];
    int ix = (int)((gx + 50.0f) * 2.0f);               // trunc((g - (bx - dx/2)) / dx)
    int iy = (int)((gy + 50.0f) * 2.0f);
    int iz = (int)((gz + 10.0f) * (1.0f / 20.0f));
    bool kept = (ix >= 0) & (ix < NXY) & (iy >= 0) & (iy < NXY) & (iz == 0);
    cellOff[bn * DHW + pt] = kept ? (ix * NXY + iy) : -1;
  }
}

// ---- kernel 3: scatter-add. One wave per point; lane l handles channels 2l, 2l+1.
__global__ void lss_scatter(const float* __restrict__ xf, const int* __restrict__ cellOff,
                            float* __restrict__ out) {
  int gt   = blockIdx.x * blockDim.x + threadIdx.x;
  int p    = gt >> 5;
  int lane = threadIdx.x & 31;
  int cell = cellOff[p];                               // wave-uniform
  if (cell < 0) return;                                // uniform branch: skip dropped points
  int b = p / (NN * DHW);
  const float2* f2 = reinterpret_cast<const float2*>(xf) + ((size_t)p * 32);
  float2 f = f2[lane];                                 // coalesced 256B per wave
  float* o = out + (size_t)b * (CC * CELLS) + (size_t)(2 * lane) * CELLS + cell;
  atomicAdd(o, f.x);                                   // L2-resident f32 atomics (41MB grid << 192MB L2)
  atomicAdd(o + CELLS, f.y);
}

extern "C" void kernel_launch(void* const* d_in, const int* in_sizes, int n_in,
                              void* d_out, int out_size, void* d_ws, size_t ws_size,
                              hipStream_t stream) {
  const float* xf         = (const float*)d_in[0];
  const float* rots       = (const float*)d_in[1];
  const float* trans      = (const float*)d_in[2];
  const float* intrins    = (const float*)d_in[3];
  const float* post_rots  = (const float*)d_in[4];
  const float* post_trans = (const float*)d_in[5];
  float* out = (float*)d_out;

  float* mats   = (float*)d_ws;                        // 24 * 32 floats = 3 KB
  int*   cellOf = (int*)((char*)d_ws + 4096);          // NPTS ints = 2.77 MB

  lss_zero   <<<OUTF / (4 * 256), 256, 0, stream>>>((float4*)out);
  lss_setup  <<<1, 32, 0, stream>>>(rots, trans, intrins, post_rots, post_trans, mats);
  lss_geom   <<<(BB * NN * (DHW / 16)) / 8, 256, 0, stream>>>(mats, cellOf);   // 5412 blocks
  lss_scatter<<<NPTS / 8, 256, 0, stream>>>(xf, cellOf, out);                  // 86592 blocks
}